// PSICHICPlusNet_77137612636205
// MI455X (gfx1250) — compile-verified
//
#include <hip/hip_runtime.h>
#include <hip/hip_bf16.h>
#include <math.h>

#define BB 16
#define NN 1024
#define FF 512
#define KK 128
#define EPSF 1e-15f

typedef __attribute__((ext_vector_type(2))) float v2f;
typedef __attribute__((ext_vector_type(8))) float v8f;

// ---------------------------------------------------------------------------
// Kernel 1: ragged -> dense scatter + row softmax.
// One block (128 threads) per ragged row t. Thread k handles s[t,k]; softmax
// via LDS tree reduce. Also copies x row (512 floats, 4 per thread).
// ---------------------------------------------------------------------------
__global__ void scatter_softmax_kernel(const float* __restrict__ x,
                                       const float* __restrict__ s,
                                       const int* __restrict__ flat_index,
                                       float* __restrict__ sm_d,  // [B*N,K]
                                       float* __restrict__ xd,    // [B*N,F]
                                       int T) {
  int t = blockIdx.x;
  if (t >= T) return;
  int tid = threadIdx.x;  // 0..127
  int row = flat_index[t];
  __shared__ float red[KK];
  float v = s[(size_t)t * KK + tid];
  red[tid] = v;
  __syncthreads();
  for (int off = 64; off > 0; off >>= 1) {
    if (tid < off) red[tid] = fmaxf(red[tid], red[tid + off]);
    __syncthreads();
  }
  float mx = red[0];
  __syncthreads();
  float e = __expf(v - mx);
  red[tid] = e;
  __syncthreads();
  for (int off = 64; off > 0; off >>= 1) {
    if (tid < off) red[tid] += red[tid + off];
    __syncthreads();
  }
  float inv = 1.0f / red[0];
  sm_d[(size_t)row * KK + tid] = e * inv;
#pragma unroll
  for (int i = tid; i < FF; i += KK)
    xd[(size_t)row * FF + i] = x[(size_t)t * FF + i];
}

// ---------------------------------------------------------------------------
// GEMM variant 1: C[M,P] = S^T * D per batch, with S:[Nd,M], D:[Nd,P] row-major.
// One wave (32 lanes) computes a 32 x 64 strip of C via V_WMMA_F32_16X16X4_F32:
// two M-tiles (a0, a1) share the four B fragments per k-step, so 8 WMMAs
// (16384 FLOP) amortize 12 b32 loads -> ~10.9 FLOP/byte from L2.
// A 16x4 fp32 layout: lanes 0-15 hold (K=0,K=1), lanes 16-31 hold (K=2,K=3).
// B 4x16 fp32 layout: symmetric. C 16x16 f32: vgpr r = row r (+8 for hi half).
// ---------------------------------------------------------------------------
__global__ __launch_bounds__(32)
void gemm_st_wmma(const float* __restrict__ S, const float* __restrict__ D,
                  float* __restrict__ C, int Nd, int M, int P) {
  const int lane = threadIdx.x;
  const int hf   = lane >> 4;   // 0/1
  const int l16  = lane & 15;
  const int ptiles = P >> 6;    // 64-wide C strips
  const int mtiles = M >> 5;    // 32-tall C strips
  int gid = blockIdx.x;
  int b  = gid / (ptiles * mtiles);
  int r  = gid % (ptiles * mtiles);
  int mt = r / ptiles;
  int pt = r % ptiles;
  const float* Sb = S + (size_t)b * Nd * M + mt * 32;
  const float* Db = D + (size_t)b * Nd * P + pt * 64;
  float* Cb = C + (size_t)b * M * P + (size_t)(mt * 32) * P + pt * 64;

  v8f acc00 = {}, acc01 = {}, acc02 = {}, acc03 = {};
  v8f acc10 = {}, acc11 = {}, acc12 = {}, acc13 = {};
  for (int n0 = 0; n0 < Nd; n0 += 4) {
    int k0 = n0 + 2 * hf;
    const float* s0 = Sb + (size_t)k0 * M + l16;
    v2f a0, a1;
    a0.x = s0[0];  a0.y = s0[M];        // rows mt*32 .. +15
    a1.x = s0[16]; a1.y = s0[M + 16];   // rows mt*32+16 .. +31
    const float* d0 = Db + (size_t)k0 * P + l16;
    const float* d1 = d0 + P;
    v2f b0, b1, b2, b3;
    b0.x = d0[0];  b0.y = d1[0];
    b1.x = d0[16]; b1.y = d1[16];
    b2.x = d0[32]; b2.y = d1[32];
    b3.x = d0[48]; b3.y = d1[48];
    acc00 = __builtin_amdgcn_wmma_f32_16x16x4_f32(false, a0, false, b0, (short)0, acc00, false, false);
    acc01 = __builtin_amdgcn_wmma_f32_16x16x4_f32(false, a0, false, b1, (short)0, acc01, false, false);
    acc02 = __builtin_amdgcn_wmma_f32_16x16x4_f32(false, a0, false, b2, (short)0, acc02, false, false);
    acc03 = __builtin_amdgcn_wmma_f32_16x16x4_f32(false, a0, false, b3, (short)0, acc03, false, false);
    acc10 = __builtin_amdgcn_wmma_f32_16x16x4_f32(false, a1, false, b0, (short)0, acc10, false, false);
    acc11 = __builtin_amdgcn_wmma_f32_16x16x4_f32(false, a1, false, b1, (short)0, acc11, false, false);
    acc12 = __builtin_amdgcn_wmma_f32_16x16x4_f32(false, a1, false, b2, (short)0, acc12, false, false);
    acc13 = __builtin_amdgcn_wmma_f32_16x16x4_f32(false, a1, false, b3, (short)0, acc13, false, false);
  }
#pragma unroll
  for (int rr = 0; rr < 8; ++rr) {
    int rowm = rr + 8 * hf;
    float* c0 = Cb + (size_t)rowm * P + l16;
    c0[0]  = acc00[rr];
    c0[16] = acc01[rr];
    c0[32] = acc02[rr];
    c0[48] = acc03[rr];
    float* c1 = c0 + (size_t)16 * P;
    c1[0]  = acc10[rr];
    c1[16] = acc11[rr];
    c1[32] = acc12[rr];
    c1[48] = acc13[rr];
  }
}

// ---------------------------------------------------------------------------
// GEMM variant 2: C[M,P] = A * D per batch, A:[M,Nd] row-major (no transpose).
// Same 32x64 per-wave tiling; A fragment loads are contiguous per lane ->
// vectorized float2 loads.
// ---------------------------------------------------------------------------
__global__ __launch_bounds__(32)
void gemm_ab_wmma(const float* __restrict__ A, const float* __restrict__ D,
                  float* __restrict__ C, int Nd, int M, int P) {
  const int lane = threadIdx.x;
  const int hf   = lane >> 4;
  const int l16  = lane & 15;
  const int ptiles = P >> 6;
  const int mtiles = M >> 5;
  int gid = blockIdx.x;
  int b  = gid / (ptiles * mtiles);
  int r  = gid % (ptiles * mtiles);
  int mt = r / ptiles;
  int pt = r % ptiles;
  const float* Ab = A + (size_t)b * M * Nd + (size_t)(mt * 32) * Nd;
  const float* Db = D + (size_t)b * Nd * P + pt * 64;
  float* Cb = C + (size_t)b * M * P + (size_t)(mt * 32) * P + pt * 64;

  v8f acc00 = {}, acc01 = {}, acc02 = {}, acc03 = {};
  v8f acc10 = {}, acc11 = {}, acc12 = {}, acc13 = {};
  for (int n0 = 0; n0 < Nd; n0 += 4) {
    int k0 = n0 + 2 * hf;
    v2f a0 = *(const v2f*)(Ab + (size_t)l16 * Nd + k0);         // k0 even -> 8B aligned
    v2f a1 = *(const v2f*)(Ab + (size_t)(l16 + 16) * Nd + k0);
    const float* d0 = Db + (size_t)k0 * P + l16;
    const float* d1 = d0 + P;
    v2f b0, b1, b2, b3;
    b0.x = d0[0];  b0.y = d1[0];
    b1.x = d0[16]; b1.y = d1[16];
    b2.x = d0[32]; b2.y = d1[32];
    b3.x = d0[48]; b3.y = d1[48];
    acc00 = __builtin_amdgcn_wmma_f32_16x16x4_f32(false, a0, false, b0, (short)0, acc00, false, false);
    acc01 = __builtin_amdgcn_wmma_f32_16x16x4_f32(false, a0, false, b1, (short)0, acc01, false, false);
    acc02 = __builtin_amdgcn_wmma_f32_16x16x4_f32(false, a0, false, b2, (short)0, acc02, false, false);
    acc03 = __builtin_amdgcn_wmma_f32_16x16x4_f32(false, a0, false, b3, (short)0, acc03, false, false);
    acc10 = __builtin_amdgcn_wmma_f32_16x16x4_f32(false, a1, false, b0, (short)0, acc10, false, false);
    acc11 = __builtin_amdgcn_wmma_f32_16x16x4_f32(false, a1, false, b1, (short)0, acc11, false, false);
    acc12 = __builtin_amdgcn_wmma_f32_16x16x4_f32(false, a1, false, b2, (short)0, acc12, false, false);
    acc13 = __builtin_amdgcn_wmma_f32_16x16x4_f32(false, a1, false, b3, (short)0, acc13, false, false);
  }
#pragma unroll
  for (int rr = 0; rr < 8; ++rr) {
    int rowm = rr + 8 * hf;
    float* c0 = Cb + (size_t)rowm * P + l16;
    c0[0]  = acc00[rr];
    c0[16] = acc01[rr];
    c0[32] = acc02[rr];
    c0[48] = acc03[rr];
    float* c1 = c0 + (size_t)16 * P;
    c1[0]  = acc10[rr];
    c1[16] = acc11[rr];
    c1[32] = acc12[rr];
    c1[48] = acc13[rr];
  }
}

// ---------------------------------------------------------------------------
// Per (b,n) row: degree d = sum_m adj[b,n,m], ssq = sum_k sm[b,n,k]^2.
// Accumulate den[b] += d*ssq (mincut denominator) and trss[b] += ssq (= tr(S^T S)).
// ---------------------------------------------------------------------------
__global__ void rowred_kernel(const float* __restrict__ adj,
                              const float* __restrict__ sm_d,
                              float* __restrict__ den, float* __restrict__ trss) {
  int bn  = blockIdx.x;      // 0..B*N-1
  int b   = bn / NN;
  int tid = threadIdx.x;     // 0..127
  __shared__ float r1[KK], r2[KK];
  const float* arow = adj + (size_t)bn * NN;
  float dsum = 0.f;
  for (int i = tid; i < NN; i += KK) dsum += arow[i];
  float sv = sm_d[(size_t)bn * KK + tid];
  r1[tid] = dsum;
  r2[tid] = sv * sv;
  __syncthreads();
  for (int off = 64; off > 0; off >>= 1) {
    if (tid < off) { r1[tid] += r1[tid + off]; r2[tid] += r2[tid + off]; }
    __syncthreads();
  }
  if (tid == 0) {
    atomicAdd(&den[b],  r1[0] * r2[0]);
    atomicAdd(&trss[b], r2[0]);
  }
}

// ||ss||_F^2 per batch.
__global__ void ssnorm_kernel(const float* __restrict__ ss, float* __restrict__ ssn2) {
  int b = blockIdx.x, tid = threadIdx.x;  // 256 threads
  __shared__ float r[256];
  const float* p = ss + (size_t)b * KK * KK;
  float acc = 0.f;
  for (int i = tid; i < KK * KK; i += 256) { float v = p[i]; acc += v * v; }
  r[tid] = acc;
  __syncthreads();
  for (int off = 128; off > 0; off >>= 1) {
    if (tid < off) r[tid] += r[tid + off];
    __syncthreads();
  }
  if (tid == 0) ssn2[b] = r[0];
}

// ---------------------------------------------------------------------------
// out_adj post-processing (in place): grab raw trace (mincut numerator),
// zero diagonal, symmetric degree-normalize with EPS.
// One block per batch, thread k owns row k.
// ---------------------------------------------------------------------------
__global__ void norm_adj_kernel(float* __restrict__ oa, float* __restrict__ num) {
  int b = blockIdx.x, tid = threadIdx.x;  // 128 threads
  float* pa = oa + (size_t)b * KK * KK;
  __shared__ float dsh[KK];
  __shared__ float diag[KK];
  float dg = pa[tid * KK + tid];
  float rs = 0.f;
  for (int l = 0; l < KK; ++l) rs += pa[tid * KK + l];
  rs -= dg;  // diagonal is zeroed before the row-sum in the reference
  dsh[tid]  = sqrtf(rs) + EPSF;
  diag[tid] = dg;
  __syncthreads();
  if (tid == 0) {
    float t = 0.f;
    for (int k = 0; k < KK; ++k) t += diag[k];
    num[b] = t;
  }
  float dk = dsh[tid];
  for (int l = 0; l < KK; ++l) {
    float v = (l == tid) ? 0.f : pa[tid * KK + l] / (dk * dsh[l]);
    pa[tid * KK + l] = v;
  }
}

// Final scalars. ortho_b = sqrt(2 - 2*tr(ss)/(||ss||_F * sqrt(K))) (exact
// closed form of ||ss/||ss|| - I/sqrt(K)||_F for any ss).
__global__ void loss_kernel(const float* __restrict__ num, const float* __restrict__ den,
                            const float* __restrict__ trss, const float* __restrict__ ssn2,
                            float* __restrict__ out_loss) {
  if (blockIdx.x == 0 && threadIdx.x == 0) {
    float mc = 0.f, ortho = 0.f;
    float sk = sqrtf((float)KK);
    for (int b = 0; b < BB; ++b) {
      mc += -(num[b] / den[b]);
      float nrm = sqrtf(ssn2[b]);
      float v = 2.f - 2.f * trss[b] / (nrm * sk);
      ortho += sqrtf(fmaxf(v, 0.f));
    }
    out_loss[0] = mc / (float)BB;
    out_loss[1] = ortho / (float)BB;
  }
}

extern "C" void kernel_launch(void* const* d_in, const int* in_sizes, int n_in,
                              void* d_out, int out_size, void* d_ws, size_t ws_size,
                              hipStream_t stream) {
  (void)n_in; (void)out_size; (void)ws_size;
  const float* x          = (const float*)d_in[0];
  const float* s          = (const float*)d_in[1];
  const float* adj        = (const float*)d_in[2];
  const int*   flat_index = (const int*)d_in[3];
  // d_in[4] (mask) is redundant: sm rows are zero wherever flat_index never writes.
  int T = in_sizes[0] / FF;

  // Output layout (concatenated, reference return order)
  float* out_sm   = (float*)d_out;                       // [B,N,K]
  float* out_feat = out_sm   + (size_t)BB * NN * KK;     // [B,K,F]
  float* out_adj  = out_feat + (size_t)BB * KK * FF;     // [B,K,K]
  float* out_loss = out_adj  + (size_t)BB * KK * KK;     // 2 scalars

  // Workspace layout
  float* xd   = (float*)d_ws;                            // [B,N,F]  32 MB
  float* sa   = xd  + (size_t)BB * NN * FF;              // [B,K,N]   8 MB
  float* ss   = sa  + (size_t)BB * KK * NN;              // [B,K,K]   1 MB
  float* den  = ss  + (size_t)BB * KK * KK;              // [B]
  float* trss = den + BB;                                // [B]
  float* num  = trss + BB;                               // [B]
  float* ssn2 = num + BB;                                // [B]

  hipMemsetAsync(out_sm, 0, sizeof(float) * (size_t)BB * NN * KK, stream);
  hipMemsetAsync(xd,     0, sizeof(float) * (size_t)BB * NN * FF, stream);
  hipMemsetAsync(den,    0, sizeof(float) * 4 * BB, stream);

  scatter_softmax_kernel<<<T, KK, 0, stream>>>(x, s, flat_index, out_sm, xd, T);

  // out = sm^T * xd        [K x N] x [N x F]
  gemm_st_wmma<<<BB * (KK / 32) * (FF / 64), 32, 0, stream>>>(out_sm, xd, out_feat, NN, KK, FF);
  // sa = sm^T * adj        [K x N] x [N x N]   (dominant GEMM)
  gemm_st_wmma<<<BB * (KK / 32) * (NN / 64), 32, 0, stream>>>(out_sm, adj, sa, NN, KK, NN);
  // ss = sm^T * sm         [K x N] x [N x K]
  gemm_st_wmma<<<BB * (KK / 32) * (KK / 64), 32, 0, stream>>>(out_sm, out_sm, ss, NN, KK, KK);
  // out_adj = sa * sm      [K x N] x [N x K]
  gemm_ab_wmma<<<BB * (KK / 32) * (KK / 64), 32, 0, stream>>>(sa, out_sm, out_adj, NN, KK, KK);

  rowred_kernel<<<BB * NN, KK, 0, stream>>>(adj, out_sm, den, trss);
  ssnorm_kernel<<<BB, 256, 0, stream>>>(ss, ssn2);
  norm_adj_kernel<<<BB, KK, 0, stream>>>(out_adj, num);
  loss_kernel<<<1, 32, 0, stream>>>(num, den, trss, ssn2, out_loss);
}